// CrossMultiHeadAttention_57526791963209
// MI455X (gfx1250) — compile-verified
//
#include <hip/hip_runtime.h>
#include <cstdint>

typedef __attribute__((ext_vector_type(2))) float v2f;
typedef __attribute__((ext_vector_type(8))) float v8f;

#define TT   26
#define NH   8
#define CC   256
#define CPH  32
#define HW_  256
#define NB   52      // B*T

__device__ __forceinline__ v8f wmma_f32(v2f a, v2f b, v8f c) {
  // V_WMMA_F32_16X16X4_F32 : D(16x16) = A(16x4) * B(4x16) + C
  return __builtin_amdgcn_wmma_f32_16x16x4_f32(false, a, false, b, (short)0, c,
                                               false, false);
}

// ---------------------------------------------------------------------------
// Kernel 1: QKV projection.  Y[n][o][p] = sum_c w_qkv[o][c] * x[n][c][p] + b[o]
// per-wave tile: M=16 (o), N=64 (p), K=256 (c), K-step 4.
// ---------------------------------------------------------------------------
__global__ void __launch_bounds__(256) qkv_gemm(const float* __restrict__ x,
                                                const float* __restrict__ wq,
                                                const float* __restrict__ bq,
                                                float* __restrict__ Y) {
  const int tid  = threadIdx.x;
  const int lane = tid & 31, wave = tid >> 5;
  const int l16  = lane & 15, g16 = lane >> 4;
  const int job   = blockIdx.x * 8 + wave;     // n*192 + mtile*4 + ptile
  const int ptile = job & 3;
  const int mtile = (job >> 2) % 48;
  const int n     = job / 192;
  const int m0    = mtile * 16;
  const int p0    = ptile * 64;
  const float* xb = x + (size_t)n * CC * HW_;

  v8f acc0 = {}, acc1 = {}, acc2 = {}, acc3 = {};
  for (int k = 0; k < CC; k += 4) {
    // A fragment: (m0+l16, k + 2*g16 + {0,1})
    const float* wp = wq + (size_t)(m0 + l16) * CC + k + 2 * g16;
    v2f a; a[0] = wp[0]; a[1] = wp[1];
    // B fragments: rows c = k + 2*g16 + {0,1}, cols p0 + 16j + l16
    const float* xp = xb + (size_t)(k + 2 * g16) * HW_ + p0 + l16;
    v2f b0; b0[0] = xp[0];  b0[1] = xp[HW_];
    v2f b1; b1[0] = xp[16]; b1[1] = xp[HW_ + 16];
    v2f b2; b2[0] = xp[32]; b2[1] = xp[HW_ + 32];
    v2f b3; b3[0] = xp[48]; b3[1] = xp[HW_ + 48];
    acc0 = wmma_f32(a, b0, acc0);
    acc1 = wmma_f32(a, b1, acc1);
    acc2 = wmma_f32(a, b2, acc2);
    acc3 = wmma_f32(a, b3, acc3);
  }
  float* yb = Y + ((size_t)n * 768 + m0) * HW_ + p0;
#pragma unroll
  for (int r = 0; r < 8; ++r) {
    const int   m    = r + 8 * g16;
    const float bias = bq[m0 + m];
    float* yr = yb + (size_t)m * HW_ + l16;
    yr[0]  = acc0[r] + bias;
    yr[16] = acc1[r] + bias;
    yr[32] = acc2[r] + bias;
    yr[48] = acc3[r] + bias;
  }
}

// ---------------------------------------------------------------------------
// Kernel 2: attention for one (b,t,head, row-half).
// Q'(256x32) @ Kg(32x1024) -> softmax -> @ Vg(1024x32).
// Channel mapping: channel = c'*8 + head (c' in [0,32)).
// ---------------------------------------------------------------------------
__global__ void __launch_bounds__(256) attn_kernel(const float* __restrict__ Y,
                                                   const int* __restrict__ accA,
                                                   float* __restrict__ O) {
  __shared__ float Vlds[1024 * 33];       // Vg staged as [key][c'], stride 33
  __shared__ float Slds[8 * 16 * 17];     // per-wave 16x16 P tile, stride 17

  const int blk  = blockIdx.x;            // 0..831
  const int half = blk & 1;
  const int bth  = blk >> 1;              // 0..415  = (b*T + t)*8 + head
  const int head = bth % NH;
  const int t    = (bth / NH) % TT;
  const int b    = bth / (NH * TT);
  const int n    = b * TT + t;

  const int tid  = threadIdx.x;
  const int lane = tid & 31, wave = tid >> 5;
  const int l16  = lane & 15, g16 = lane >> 4;

  int tg[4];
#pragma unroll
  for (int g = 0; g < 4; ++g) tg[g] = accA[t * 4 + g];

  // ---- stage Vg into LDS:  Vlds[key*33 + c'] = V[b, tg[key/256], c'*8+head, key%256]
  for (int i = tid; i < 1024 * 32; i += 256) {
    const int cp  = i >> 10;        // c' in [0,32)
    const int key = i & 1023;
    const int gg  = key >> 8;
    const int pp  = key & 255;
    Vlds[key * 33 + cp] =
        Y[((size_t)(b * TT + tg[gg]) * 3 * CC + 2 * CC + cp * NH + head) * HW_ + pp];
  }
  __syncthreads();

  const int m0 = half * 128 + wave * 16;            // this wave's 16 p-rows
  const float* Qbase = Y + (size_t)n * 3 * CC * HW_; // q section at channel offset 0

  // ---- Q A-fragments (reused across all 64 key tiles): c' = 4s + 2*g16 + {0,1}
  v2f aq[8];
#pragma unroll
  for (int s = 0; s < 8; ++s) {
    const int cp = 4 * s + 2 * g16;
    const float* qp = Qbase + (size_t)(cp * NH + head) * HW_ + m0 + l16;
    aq[s][0] = qp[0];
    aq[s][1] = qp[NH * HW_];     // next c' -> channel + 8
  }

  v8f o0 = {}, o1 = {};
  float rs[8];
#pragma unroll
  for (int r = 0; r < 8; ++r) rs[r] = 0.0f;
  const float scale = 0.17677669529663687f;  // 1/sqrt(32)
  float* Sw = &Slds[wave * 16 * 17];

  for (int kt = 0; kt < 64; ++kt) {
    const int gg    = kt >> 4;
    const int pbase = (kt & 15) * 16;
    const float* Kb =
        Y + ((size_t)(b * TT + tg[gg]) * 3 * CC + CC) * HW_;  // k section

    // S tile (16 p-rows x 16 keys), K over c' (32) in 8 steps
    v8f sacc = {};
#pragma unroll
    for (int s = 0; s < 8; ++s) {
      const int cp = 4 * s + 2 * g16;
      const float* kp = Kb + (size_t)(cp * NH + head) * HW_ + pbase + l16;
      v2f bk; bk[0] = kp[0]; bk[1] = kp[NH * HW_];
      sacc = wmma_f32(aq[s], bk, sacc);
    }

    // P = exp(S*scale); accumulate row sums; spill P tile through LDS
#pragma unroll
    for (int r = 0; r < 8; ++r) {
      const float p = __expf(sacc[r] * scale);
      rs[r] += p;
      Sw[(r + 8 * g16) * 17 + l16] = p;   // row (r+8g), key-col l16
    }
    asm volatile("s_wait_dscnt 0" ::: "memory");  // in-wave DS RAW ordering

    // O += P(16x16) @ Vtile(16x32): K over 16 keys in 4 steps, 2 N-tiles
#pragma unroll
    for (int s = 0; s < 4; ++s) {
      const int krow = 4 * s + 2 * g16;            // local key this lane holds
      v2f ap;
      ap[0] = Sw[l16 * 17 + krow];                  // P[m=l16][krow]
      ap[1] = Sw[l16 * 17 + krow + 1];
      const int key = kt * 16 + krow;
      v2f bv0, bv1;
      bv0[0] = Vlds[key * 33 + l16];                // V[key][c'=l16]
      bv0[1] = Vlds[(key + 1) * 33 + l16];
      bv1[0] = Vlds[key * 33 + 16 + l16];           // c' = 16+l16
      bv1[1] = Vlds[(key + 1) * 33 + 16 + l16];
      o0 = wmma_f32(ap, bv0, o0);
      o1 = wmma_f32(ap, bv1, o1);
    }
  }

  // reduce row sums across the 16-lane fragment group (rows r+8g stay coherent)
#pragma unroll
  for (int r = 0; r < 8; ++r) {
    float v = rs[r];
    v += __shfl_xor(v, 1, 32);
    v += __shfl_xor(v, 2, 32);
    v += __shfl_xor(v, 4, 32);
    v += __shfl_xor(v, 8, 32);
    rs[r] = 1.0f / v;
  }

  // store O: row p = m0 + r + 8*g16, channel = c'*8 + head
  float* Ob = O + (size_t)n * CC * HW_;
#pragma unroll
  for (int r = 0; r < 8; ++r) {
    const int p = m0 + r + 8 * g16;
    Ob[(size_t)(l16 * NH + head) * HW_ + p]        = o0[r] * rs[r];
    Ob[(size_t)((16 + l16) * NH + head) * HW_ + p] = o1[r] * rs[r];
  }
}

// ---------------------------------------------------------------------------
// Kernel 3: output projection. out[n][o][p] = sum_c w_out[o][c]*O[n][c][p] + b[o]
// ---------------------------------------------------------------------------
__global__ void __launch_bounds__(256) proj_gemm(const float* __restrict__ Oin,
                                                 const float* __restrict__ wo,
                                                 const float* __restrict__ bo,
                                                 float* __restrict__ out) {
  const int tid  = threadIdx.x;
  const int lane = tid & 31, wave = tid >> 5;
  const int l16  = lane & 15, g16 = lane >> 4;
  const int job   = blockIdx.x * 8 + wave;   // n*64 + mtile*4 + ptile
  const int ptile = job & 3;
  const int mtile = (job >> 2) & 15;
  const int n     = job >> 6;
  const int m0    = mtile * 16;
  const int p0    = ptile * 64;
  const float* ob = Oin + (size_t)n * CC * HW_;

  v8f acc0 = {}, acc1 = {}, acc2 = {}, acc3 = {};
  for (int k = 0; k < CC; k += 4) {
    const float* wp = wo + (size_t)(m0 + l16) * CC + k + 2 * g16;
    v2f a; a[0] = wp[0]; a[1] = wp[1];
    const float* xp = ob + (size_t)(k + 2 * g16) * HW_ + p0 + l16;
    v2f b0; b0[0] = xp[0];  b0[1] = xp[HW_];
    v2f b1; b1[0] = xp[16]; b1[1] = xp[HW_ + 16];
    v2f b2; b2[0] = xp[32]; b2[1] = xp[HW_ + 32];
    v2f b3; b3[0] = xp[48]; b3[1] = xp[HW_ + 48];
    acc0 = wmma_f32(a, b0, acc0);
    acc1 = wmma_f32(a, b1, acc1);
    acc2 = wmma_f32(a, b2, acc2);
    acc3 = wmma_f32(a, b3, acc3);
  }
  float* yb = out + ((size_t)n * CC + m0) * HW_ + p0;
#pragma unroll
  for (int r = 0; r < 8; ++r) {
    const int   m    = r + 8 * g16;
    const float bias = bo[m0 + m];
    float* yr = yb + (size_t)m * HW_ + l16;
    yr[0]  = acc0[r] + bias;
    yr[16] = acc1[r] + bias;
    yr[32] = acc2[r] + bias;
    yr[48] = acc3[r] + bias;
  }
}

// ---------------------------------------------------------------------------
extern "C" void kernel_launch(void* const* d_in, const int* in_sizes, int n_in,
                              void* d_out, int out_size, void* d_ws, size_t ws_size,
                              hipStream_t stream) {
  const float* x      = (const float*)d_in[0];  // (52, 256, 16, 16)
  const float* w_qkv  = (const float*)d_in[1];  // (768, 256)
  const float* b_qkv  = (const float*)d_in[2];  // (768)
  const float* w_out  = (const float*)d_in[3];  // (256, 256)
  const float* b_out  = (const float*)d_in[4];  // (256)
  const int*   accA   = (const int*)d_in[5];    // (26, 4)
  float* outp = (float*)d_out;

  float* Y = (float*)d_ws;                 // QKV: 52 * 768 * 256 floats
  float* O = Y + (size_t)NB * 768 * HW_;   // attention out: 52 * 256 * 256 floats

  // 1) QKV projection: 52*48*4 wave-jobs / 8 waves = 1248 blocks
  qkv_gemm<<<1248, 256, 0, stream>>>(x, w_qkv, b_qkv, Y);
  // 2) attention: 416 (b,t,head) * 2 row-halves
  attn_kernel<<<832, 256, 0, stream>>>(Y, accA, O);
  // 3) output projection: 52*16*4 / 8 = 416 blocks
  proj_gemm<<<416, 256, 0, stream>>>(O, w_out, b_out, outp);
}